// ODEfunc_39419209842964
// MI455X (gfx1250) — compile-verified
//
#include <hip/hip_runtime.h>

typedef __attribute__((ext_vector_type(2))) float v2f;
typedef __attribute__((ext_vector_type(8))) float v8f;

#define DD    32
#define PREV  64
#define HIDN  128
#define HDIM  64
#define NW    ((DD + 1) * PREV)   // 2112
#define YCOLS (DD + 1)            // 33
#define BATCH 8192

#define WAVES 4
#define LSTR  132                 // padded LDS row stride (floats)

// ---------------------------------------------------------------------------
// Prep kernel: hypernetwork -> w1 (32x64), b1 (64), and the batch-independent
// divergence vector M[h] = sum_p bw[h,p] * (w1^T @ w2)[p,h].
// ws layout (floats): [0..2047]=w1, [2048..2111]=b1, [2112..2239]=M
// ---------------------------------------------------------------------------
__global__ __launch_bounds__(256) void prep_kernel(
    const float* __restrict__ t,  const float* __restrict__ hw1,
    const float* __restrict__ hb1, const float* __restrict__ hw2,
    const float* __restrict__ hb2, const float* __restrict__ bw,
    const float* __restrict__ w2,  float* __restrict__ ws)
{
    __shared__ float sh_hcode[HDIM];
    __shared__ float sh_w[NW];
    const int tid = threadIdx.x;

    if (tid < HDIM)
        sh_hcode[tid] = tanhf(t[0] * hw1[tid] + hb1[tid]);
    __syncthreads();

    for (int j = tid; j < NW; j += 256) {
        float acc = hb2[j];
        #pragma unroll 8
        for (int k = 0; k < HDIM; ++k)
            acc += hw2[j * HDIM + k] * sh_hcode[k];
        sh_w[j] = acc;
        ws[j]   = acc;
    }
    __syncthreads();

    if (tid < HIDN) {
        float m = 0.f;
        for (int p = 0; p < PREV; ++p) {
            float c = 0.f;
            #pragma unroll 8
            for (int d = 0; d < DD; ++d)
                c += sh_w[d * PREV + p] * w2[d * HIDN + tid];
            m += bw[tid * PREV + p] * c;
        }
        ws[NW + tid] = m;
    }
}

// ---------------------------------------------------------------------------
// Main kernel: one wave32 per 16-row batch tile; three chained fp32 WMMA
// GEMMs (16x16x4) with LDS re-marshaling between C-layout and A-layout.
// ---------------------------------------------------------------------------
__global__ __launch_bounds__(WAVES * 32) void odefunc_kernel(
    const float* __restrict__ y,  const float* __restrict__ ws,
    const float* __restrict__ bw, const float* __restrict__ bb,
    const float* __restrict__ w2, const float* __restrict__ b2,
    float* __restrict__ out)
{
    __shared__ float lds[WAVES * 16 * LSTR];

    const float* w1 = ws;               // 32x64 row-major
    const float* b1 = ws + DD * PREV;   // 64
    const float* Mv = ws + NW;          // 128

    const int tid  = threadIdx.x;
    const int wave = tid >> 5;
    const int lane = tid & 31;
    const int hl   = lane & 15;                 // lane within half-wave
    const int half = lane >> 4;                 // 0 or 1
    const int ksel = half << 1;                 // K sub-offset: 0 or 2
    const int mofs = half << 3;                 // C/D row offset: 0 or 8
    float* sm = &lds[wave * 16 * LSTR];

    const int rowbase = (blockIdx.x * WAVES + wave) * 16;

    // ---- GEMM1: z = x @ w1 + b1  (16x64, K=32) -> 4 N-tiles, 8 K-steps
    v8f zacc[4];
    #pragma unroll
    for (int j = 0; j < 4; ++j) {
        const float bias = b1[j * 16 + hl];
        #pragma unroll
        for (int r = 0; r < 8; ++r) zacc[j][r] = bias;
    }
    v2f afrag[8];
    {
        const float* xrow = y + (size_t)(rowbase + hl) * YCOLS;
        #pragma unroll
        for (int k = 0; k < 8; ++k) {
            const int col = k * 4 + ksel;
            afrag[k].x = xrow[col];
            afrag[k].y = xrow[col + 1];
        }
    }
    #pragma unroll
    for (int j = 0; j < 4; ++j) {
        const int n = j * 16 + hl;
        #pragma unroll
        for (int k = 0; k < 8; ++k) {
            v2f b;
            b.x = w1[(k * 4 + ksel)     * PREV + n];
            b.y = w1[(k * 4 + ksel + 1) * PREV + n];
            zacc[j] = __builtin_amdgcn_wmma_f32_16x16x4_f32(
                false, afrag[k], false, b, (short)0, zacc[j], false, false);
        }
    }

    // stage z into LDS (C-layout -> memory row-major)
    #pragma unroll
    for (int j = 0; j < 4; ++j)
        #pragma unroll
        for (int r = 0; r < 8; ++r)
            sm[(r + mofs) * LSTR + j * 16 + hl] = zacc[j][r];
    asm volatile("s_wait_dscnt 0" ::: "memory");

    // ---- GEMM2: s = z @ bw^T + bb  (16x128, K=64) -> 8 N-tiles, 16 K-steps
    v8f sacc[8];
    #pragma unroll
    for (int j = 0; j < 8; ++j) {
        const float bias = bb[j * 16 + hl];
        #pragma unroll
        for (int r = 0; r < 8; ++r) sacc[j][r] = bias;
    }
    v2f zfrag[16];
    #pragma unroll
    for (int k = 0; k < 16; ++k) {
        const int col = k * 4 + ksel;
        zfrag[k].x = sm[hl * LSTR + col];
        zfrag[k].y = sm[hl * LSTR + col + 1];
    }
    #pragma unroll
    for (int j = 0; j < 8; ++j) {
        const float* bwrow = bw + (size_t)(j * 16 + hl) * PREV;  // bw^T[k][h]
        #pragma unroll
        for (int k = 0; k < 16; ++k) {
            v2f b;
            b.x = bwrow[k * 4 + ksel];
            b.y = bwrow[k * 4 + ksel + 1];
            sacc[j] = __builtin_amdgcn_wmma_f32_16x16x4_f32(
                false, zfrag[k], false, b, (short)0, sacc[j], false, false);
        }
    }

    // ---- a = tanh(s); divergence partials; stage a into LDS
    float divp[8];
    #pragma unroll
    for (int r = 0; r < 8; ++r) divp[r] = 0.f;
    #pragma unroll
    for (int j = 0; j < 8; ++j) {
        const int h = j * 16 + hl;
        const float mh = Mv[h];
        #pragma unroll
        for (int r = 0; r < 8; ++r) {
            const float a = tanhf(sacc[j][r]);
            divp[r] += (1.f - a * a) * mh;
            sm[(r + mofs) * LSTR + h] = a;
        }
    }
    asm volatile("s_wait_dscnt 0" ::: "memory");

    // ---- GEMM3: dx = a @ w2^T + b2  (16x32, K=128) -> 2 N-tiles, 32 K-steps
    v8f dacc[2];
    #pragma unroll
    for (int j = 0; j < 2; ++j) {
        const float bias = b2[j * 16 + hl];
        #pragma unroll
        for (int r = 0; r < 8; ++r) dacc[j][r] = bias;
    }
    #pragma unroll
    for (int j = 0; j < 2; ++j) {
        const int n = j * 16 + hl;                  // output column d
        #pragma unroll
        for (int k = 0; k < 32; ++k) {
            const int col = k * 4 + ksel;           // K index h
            v2f a, b;
            a.x = sm[hl * LSTR + col];
            a.y = sm[hl * LSTR + col + 1];
            b.x = w2[n * HIDN + col];               // w2^T[h][d] = w2[d][h]
            b.y = w2[n * HIDN + col + 1];
            dacc[j] = __builtin_amdgcn_wmma_f32_16x16x4_f32(
                false, a, false, b, (short)0, dacc[j], false, false);
        }
    }

    // reduce div across each 16-lane half (rows 0-7 in lanes 0-15, 8-15 in 16-31)
    #pragma unroll
    for (int s = 1; s < 16; s <<= 1)
        #pragma unroll
        for (int r = 0; r < 8; ++r)
            divp[r] += __shfl_xor(divp[r], s, 32);

    // ---- write out: cols 0..31 = dx, col 32 = -div
    #pragma unroll
    for (int j = 0; j < 2; ++j) {
        const int n = j * 16 + hl;
        #pragma unroll
        for (int r = 0; r < 8; ++r)
            out[(size_t)(rowbase + r + mofs) * YCOLS + n] = dacc[j][r];
    }
    if (hl == 0) {
        #pragma unroll
        for (int r = 0; r < 8; ++r)
            out[(size_t)(rowbase + r + mofs) * YCOLS + DD] = -divp[r];
    }
}

// ---------------------------------------------------------------------------
extern "C" void kernel_launch(void* const* d_in, const int* in_sizes, int n_in,
                              void* d_out, int out_size, void* d_ws, size_t ws_size,
                              hipStream_t stream) {
    const float* y   = (const float*)d_in[0];
    const float* t   = (const float*)d_in[1];
    const float* hw1 = (const float*)d_in[2];
    const float* hb1 = (const float*)d_in[3];
    const float* hw2 = (const float*)d_in[4];
    const float* hb2 = (const float*)d_in[5];
    const float* bw  = (const float*)d_in[6];
    const float* bb  = (const float*)d_in[7];
    const float* w2  = (const float*)d_in[8];
    const float* b2  = (const float*)d_in[9];
    float* out = (float*)d_out;
    float* ws  = (float*)d_ws;

    prep_kernel<<<1, 256, 0, stream>>>(t, hw1, hb1, hw2, hb2, bw, w2, ws);

    const int tiles  = BATCH / 16;           // 512 wave-tiles
    const int blocks = tiles / WAVES;        // 128 blocks of 4 waves
    odefunc_kernel<<<blocks, WAVES * 32, 0, stream>>>(y, ws, bw, bb, w2, b2, out);
}